// StochasticPool2DLayer_43044162241228
// MI455X (gfx1250) — compile-verified
//
#include <hip/hip_runtime.h>

// -----------------------------------------------------------------------------
// Stochastic 2x2 pool (eval mode), x:(64,256,112,112) f32 -> out:(64,256,56,56)
// Memory-bound (AI ~0.75 flop/byte -> HBM floor ~44us @ 23.3 TB/s).
// Strategy: stream HBM -> LDS with gfx1250 async copies (4-deep pipeline),
// compute from LDS (ds_load_b128, conflict-free), non-temporal b64 stores.
// -----------------------------------------------------------------------------

typedef float v4f __attribute__((ext_vector_type(4)));
typedef float v2f __attribute__((ext_vector_type(2)));
typedef int   v4i __attribute__((ext_vector_type(4)));

#define N_PLANES   16384      // 64 * 256
#define IN_PLANE   12544      // 112 * 112
#define OUT_PLANE  3136       // 56 * 56
#define W_IN       112
#define GROUPS     7          // groups per plane
#define GRP_IN     1792       // 16 input rows * 112 floats (contiguous)
#define GRP_OUT    448        // 8 output rows * 56
#define THREADS    224        // 7 wave32
#define NBUF       4

#if defined(__HIP_DEVICE_COMPILE__) && __has_builtin(__builtin_amdgcn_global_load_async_to_lds_b128)
#define USE_ASYNC_LDS 1
#else
#define USE_ASYNC_LDS 0
#endif

#if USE_ASYNC_LDS
typedef __attribute__((address_space(1))) v4i* gv4i_p;   // global int4*
typedef __attribute__((address_space(3))) v4i* lv4i_p;   // LDS int4*

__device__ __forceinline__ void async_copy16(const float* g, float* l) {
  // (global src, lds dst, imm offset, imm cpol)
  __builtin_amdgcn_global_load_async_to_lds_b128((gv4i_p)g, (lv4i_p)l, 0, 0);
}

template <int N>
__device__ __forceinline__ void wait_asynccnt() {
#if __has_builtin(__builtin_amdgcn_s_wait_asynccnt)
  __builtin_amdgcn_s_wait_asynccnt(N);
#else
  asm volatile("s_wait_asynccnt %0" ::"i"(N) : "memory");
#endif
}
#endif  // USE_ASYNC_LDS

// two adjacent outputs from one 2x4 input patch (top/bot rows)
__device__ __forceinline__ v2f pool2(v4f top, v4f bot) {
  float a0 = fmaxf(top.x, 0.0f), b0 = fmaxf(top.y, 0.0f);
  float c0 = fmaxf(bot.x, 0.0f), d0 = fmaxf(bot.y, 0.0f);
  float a1 = fmaxf(top.z, 0.0f), b1 = fmaxf(top.w, 0.0f);
  float c1 = fmaxf(bot.z, 0.0f), d1 = fmaxf(bot.w, 0.0f);
  float s0 = (a0 + b0) + (c0 + d0);
  float s1 = (a1 + b1) + (c1 + d1);
  float n0 = (a0 * a0 + b0 * b0) + (c0 * c0 + d0 * d0);
  float n1 = (a1 * a1 + b1 * b1) + (c1 * c1 + d1 * d1);
  v2f r;
  r.x = (s0 > 0.0f) ? n0 / s0 : 0.0f;
  r.y = (s1 > 0.0f) ? n1 / s1 : 0.0f;
  return r;
}

__global__ __launch_bounds__(THREADS) void stochpool_kernel(
    const float* __restrict__ x, float* __restrict__ out) {
  const int i = threadIdx.x;                 // 0..223
  const int plane = blockIdx.x;              // 0..16383
  const float* src = x + (size_t)plane * IN_PLANE;
  float* dst = out + (size_t)plane * OUT_PLANE;

#if USE_ASYNC_LDS
  __shared__ __attribute__((aligned(16))) float tile[NBUF][GRP_IN];

  // Preload groups 0..2 (2 x b128 async per thread per group)
  #pragma unroll
  for (int p = 0; p < 3; ++p) {
    const float* gs = src + p * GRP_IN;
    async_copy16(gs + i * 4,             &tile[p][i * 4]);
    async_copy16(gs + (i + THREADS) * 4, &tile[p][(i + THREADS) * 4]);
  }

  #pragma unroll
  for (int g = 0; g < GROUPS; ++g) {
    if (g + 3 < GROUPS) {
      const int p = (g + 3) & (NBUF - 1);
      const float* gs = src + (g + 3) * GRP_IN;
      async_copy16(gs + i * 4,             &tile[p][i * 4]);
      async_copy16(gs + (i + THREADS) * 4, &tile[p][(i + THREADS) * 4]);
    }
    // wait until group g's copies complete (2 async instr per group per wave)
    if (g <= 3)      wait_asynccnt<6>();
    else if (g == 4) wait_asynccnt<4>();
    else if (g == 5) wait_asynccnt<2>();
    else             wait_asynccnt<0>();
    __syncthreads();

    const float* t = tile[g & (NBUF - 1)];
    const int o = 2 * i;             // output pair index within group (0..446)
    const int r = o / 56;            // output row in group (0..7)
    const int wo = o % 56;           // even
    const int base = r * 224 + 2 * wo;   // 16B-aligned LDS index
    v4f top = *(const v4f*)(t + base);
    v4f bot = *(const v4f*)(t + base + W_IN);
    v2f res = pool2(top, bot);
    __builtin_nontemporal_store(res, (v2f*)(dst + g * GRP_OUT + o));
    __syncthreads();  // tile[g&3] will be overwritten at iteration g+1
  }
#else
  // Fallback: same tiling, direct NT b128 loads (no LDS staging)
  const int o = 2 * i;
  const int r = o / 56;
  const int wo = o % 56;
  #pragma unroll
  for (int g = 0; g < GROUPS; ++g) {
    const float* rowp = src + (size_t)(16 * g + 2 * r) * W_IN + 2 * wo;
    v4f top = __builtin_nontemporal_load((const v4f*)rowp);
    v4f bot = __builtin_nontemporal_load((const v4f*)(rowp + W_IN));
    v2f res = pool2(top, bot);
    __builtin_nontemporal_store(res, (v2f*)(dst + g * GRP_OUT + o));
  }
#endif
}

extern "C" void kernel_launch(void* const* d_in, const int* in_sizes, int n_in,
                              void* d_out, int out_size, void* d_ws, size_t ws_size,
                              hipStream_t stream) {
  (void)in_sizes; (void)n_in; (void)out_size; (void)d_ws; (void)ws_size;
  const float* x = (const float*)d_in[0];
  float* out = (float*)d_out;
  stochpool_kernel<<<N_PLANES, THREADS, 0, stream>>>(x, out);
}